// SpeciesDetector_87024627351841
// MI455X (gfx1250) — compile-verified
//
#include <hip/hip_runtime.h>
#include <math.h>

typedef __attribute__((ext_vector_type(2))) float v2f;
typedef __attribute__((ext_vector_type(8))) float v8f;

#define TAU_INV (1.0f / 5.0f)
#define NN 1024
#define MM 256

__device__ __forceinline__ int lane_id() { return (int)(threadIdx.x & 31); }

// ---- V_WMMA_F32_16X16X4_F32 fragment helpers -------------------------------
// A (16x4 chunk c of row-major 16x16 X): lanes 0-15 rows, VGPR0/1 = K(2h),K(2h+1)
__device__ __forceinline__ v2f afrag(const float* X, int c) {
  int l = lane_id(), h = l >> 4, m = l & 15;
  int k0 = 4 * c + 2 * h;
  v2f r; r[0] = X[m * 16 + k0]; r[1] = X[m * 16 + k0 + 1];
  return r;
}
// B (4x16 chunk c of row-major 16x16 Y): lanes = columns, VGPRj = K(j+2h)
__device__ __forceinline__ v2f bfrag(const float* Y, int c) {
  int l = lane_id(), h = l >> 4, n = l & 15;
  int k0 = 4 * c + 2 * h;
  v2f r; r[0] = Y[k0 * 16 + n]; r[1] = Y[(k0 + 1) * 16 + n];
  return r;
}
// B fragment of Y^T (Y row-major 16x16): B[k][n] = Y[n][k]
__device__ __forceinline__ v2f btfrag(const float* Y, int c) {
  int l = lane_id(), h = l >> 4, n = l & 15;
  int k0 = 4 * c + 2 * h;
  v2f r; r[0] = Y[n * 16 + k0]; r[1] = Y[n * 16 + k0 + 1];
  return r;
}
// C/D layout store: VGPR j -> row (j + 8h), col = lane&15
__device__ __forceinline__ void store_c(float* D, v8f c) {
  int l = lane_id(), h = l >> 4, n = l & 15;
#pragma unroll
  for (int j = 0; j < 8; ++j) D[(j + 8 * h) * 16 + n] = c[j];
}

// ---- prep kernel 1: invert each omega_b[m] (Gauss-Jordan, row per lane) ----
__global__ __launch_bounds__(256) void invert_omega_b_kernel(
    const float* __restrict__ omega_b, float* __restrict__ obinv) {
  const int wave = threadIdx.x >> 5;
  const int lane = threadIdx.x & 31;
  const int m = blockIdx.x * 8 + wave;
  const int i = lane & 15;
  float a[16], e[16];
#pragma unroll
  for (int k = 0; k < 16; ++k) {
    a[k] = omega_b[m * 256 + i * 16 + k];
    e[k] = (k == i) ? 1.0f : 0.0f;
  }
#pragma unroll
  for (int j = 0; j < 16; ++j) {
    float p = __shfl(a[j], j, 32);
    float inv = 1.0f / p;
    float f = (i == j) ? 0.0f : a[j] * inv;
#pragma unroll
    for (int k = 0; k < 16; ++k) {
      float ajk = __shfl(a[k], j, 32);
      float ejk = __shfl(e[k], j, 32);
      a[k] -= f * ajk;
      e[k] -= f * ejk;
    }
    if (i == j) {
#pragma unroll
      for (int k = 0; k < 16; ++k) { a[k] *= inv; e[k] *= inv; }
    }
  }
  if (lane < 16) {
#pragma unroll
    for (int k = 0; k < 16; ++k) obinv[m * 256 + i * 16 + k] = e[k];
  }
}

// ---- prep kernel 2: Cholesky of each sigma_a[n] -> La, logdet_a ------------
__global__ __launch_bounds__(256) void chol_sigma_a_kernel(
    const float* __restrict__ sigma_a, float* __restrict__ La,
    float* __restrict__ lda) {
  const int wave = threadIdx.x >> 5;
  const int lane = threadIdx.x & 31;
  const int n = blockIdx.x * 8 + wave;
  const int i = lane & 15;
  float a[16];
#pragma unroll
  for (int k = 0; k < 16; ++k) a[k] = sigma_a[n * 256 + i * 16 + k];
  float logdet = 0.0f;
#pragma unroll
  for (int j = 0; j < 16; ++j) {
    float s = a[j];
#pragma unroll
    for (int k = 0; k < j; ++k) s -= a[k] * __shfl(a[k], j, 32);
    float sj = __shfl(s, j, 32);
    float d = sqrtf(sj);
    logdet += logf(sj);
    float rd = 1.0f / d;
    a[j] = (i == j) ? d : s * rd;
  }
  if (lane < 16) {
#pragma unroll
    for (int k = 0; k < 16; ++k)
      La[n * 256 + i * 16 + k] = (k <= i) ? a[k] : 0.0f;
  }
  if (lane == 0) lda[n] = logdet;
}

// ---- main kernel: one wave per (n,m) pair ----------------------------------
__global__ __launch_bounds__(256) void species_pair_kernel(
    const float* __restrict__ mu_a, const float* __restrict__ omega_a,
    const float* __restrict__ mu_b, const float* __restrict__ sigma_b,
    const float* __restrict__ obinv, const float* __restrict__ La,
    const float* __restrict__ logdet_a, float* __restrict__ out) {
  __shared__ float lds[8][512];
  const int wave = threadIdx.x >> 5;
  const int lane = threadIdx.x & 31;
  float* Tl = &lds[wave][0];    // transport matrix T
  float* Sl = &lds[wave][256];  // scratch: S then sigma_b_t

  const int pair = blockIdx.x * 8 + wave;
  const int n = pair >> 8;          // pair / 256
  const int m = pair & (MM - 1);

  const float* Oa  = omega_a + n * 256;
  const float* Obi = obinv   + m * 256;
  const float* Sb  = sigma_b + m * 256;

  // T = omega_a[n] * omega_b_inv[m]   (4x v_wmma_f32_16x16x4_f32)
  v8f acc = {};
#pragma unroll
  for (int c4 = 0; c4 < 4; ++c4)
    acc = __builtin_amdgcn_wmma_f32_16x16x4_f32(
        false, afrag(Oa, c4), false, bfrag(Obi, c4), (short)0, acc, false, false);
  store_c(Tl, acc);
  __builtin_amdgcn_wave_barrier();

  // S = T * sigma_b[m]
  v8f acc2 = {};
#pragma unroll
  for (int c4 = 0; c4 < 4; ++c4)
    acc2 = __builtin_amdgcn_wmma_f32_16x16x4_f32(
        false, afrag(Tl, c4), false, bfrag(Sb, c4), (short)0, acc2, false, false);
  store_c(Sl, acc2);
  __builtin_amdgcn_wave_barrier();

  // sigma_b_t = S * T^T
  v8f acc3 = {};
#pragma unroll
  for (int c4 = 0; c4 < 4; ++c4)
    acc3 = __builtin_amdgcn_wmma_f32_16x16x4_f32(
        false, afrag(Sl, c4), false, btfrag(Tl, c4), (short)0, acc3, false, false);
  __builtin_amdgcn_wave_barrier();
  store_c(Sl, acc3);  // Sl now holds sigma_b_t
  __builtin_amdgcn_wave_barrier();

  const int i = lane & 15;  // lanes 16-31 mirror rows/cols 0-15 (benign)

  // mu_b_t and diff (row i of T dot mu_b)
  float mbt = 0.0f;
#pragma unroll
  for (int k = 0; k < 16; ++k) mbt = fmaf(Tl[i * 16 + k], mu_b[m * 16 + k], mbt);
  float diff = mu_a[n * 16 + i] - mbt;

  // load row i of sigma_b_t into registers
  float a[16];
#pragma unroll
  for (int k = 0; k < 16; ++k) a[k] = Sl[i * 16 + k];

  // in-register Cholesky (row per lane, shuffle broadcasts)
  float logdet_b = 0.0f;
#pragma unroll
  for (int j = 0; j < 16; ++j) {
    float s = a[j];
#pragma unroll
    for (int k = 0; k < j; ++k) s -= a[k] * __shfl(a[k], j, 32);
    float sj = __shfl(s, j, 32);     // diag pivot (uniform)
    float d = sqrtf(sj);
    logdet_b += logf(sj);            // = 2*log(d), wave-uniform
    float rd = 1.0f / d;
    a[j] = (i == j) ? d : s * rd;
  }

  // trace = || L^-1 * La ||_F^2  : forward solve, lane = column of La
  float b[16];
#pragma unroll
  for (int j = 0; j < 16; ++j) b[j] = La[n * 256 + j * 16 + i];
  float w[16];
  float tr = 0.0f;
#pragma unroll
  for (int j = 0; j < 16; ++j) {
    float t = b[j];
#pragma unroll
    for (int k = 0; k < j; ++k) t -= __shfl(a[k], j, 32) * w[k];
    float dj = __shfl(a[j], j, 32);
    w[j] = t / dj;
    tr = fmaf(w[j], w[j], tr);
  }
#pragma unroll
  for (int off = 16; off >= 1; off >>= 1) tr += __shfl_xor(tr, off, 32);
  tr *= 0.5f;  // lanes 16-31 duplicated columns 0-15

  // quad = || L^-1 diff ||^2 : forward solve on the vector
  float r = diff, quad = 0.0f;
#pragma unroll
  for (int j = 0; j < 16; ++j) {
    float dj = __shfl(a[j], j, 32);
    float yj = __shfl(r, j, 32) / dj;  // uniform
    quad = fmaf(yj, yj, quad);
    r -= a[j] * yj;
  }

  float kl = 0.5f * (tr + quad - 16.0f + logdet_b - logdet_a[n]);
  if (lane == 0)
    out[n * MM + m] = 1.0f / (1.0f + expf(kl * TAU_INV));
}

// ---- launch ----------------------------------------------------------------
extern "C" void kernel_launch(void* const* d_in, const int* in_sizes, int n_in,
                              void* d_out, int out_size, void* d_ws, size_t ws_size,
                              hipStream_t stream) {
  (void)in_sizes; (void)n_in; (void)out_size; (void)ws_size;
  const float* mu_a    = (const float*)d_in[0];
  const float* sigma_a = (const float*)d_in[1];
  const float* omega_a = (const float*)d_in[2];
  const float* mu_b    = (const float*)d_in[3];
  const float* sigma_b = (const float*)d_in[4];
  const float* omega_b = (const float*)d_in[5];

  float* ws    = (float*)d_ws;
  float* obinv = ws;                           // 256*256 floats
  float* La    = ws + 256 * 256;               // 1024*256 floats
  float* lda   = ws + 256 * 256 + 1024 * 256;  // 1024 floats

  invert_omega_b_kernel<<<MM / 8, 256, 0, stream>>>(omega_b, obinv);
  chol_sigma_a_kernel<<<NN / 8, 256, 0, stream>>>(sigma_a, La, lda);
  species_pair_kernel<<<(NN * MM) / 8, 256, 0, stream>>>(
      mu_a, omega_a, mu_b, sigma_b, obinv, La, lda, (float*)d_out);
}